// GCNR_41961830482650
// MI455X (gfx1250) — compile-verified
//
#include <hip/hip_runtime.h>
#include <hip/hip_bf16.h>
#include <math.h>

#define NUM_ROI 200

typedef __attribute__((ext_vector_type(2))) float v2f;
typedef __attribute__((ext_vector_type(8))) float v8f;

__device__ __forceinline__ float mishf(float x) {
    float sp = (x > 20.0f) ? x : log1pf(expf(x));
    return x * tanhf(sp);
}

// ---------------- utility ----------------
__global__ void k_zero(float* __restrict__ p, long n) {
    long i = (long)blockIdx.x * blockDim.x + threadIdx.x;
    if (i < n) p[i] = 0.0f;
}

// One-time: int64 edge index -> int32 row/col arrays
__global__ void k_prep_idx(const long long* __restrict__ ei, int* __restrict__ row32,
                           int* __restrict__ col32, int E) {
    long i = (long)blockIdx.x * blockDim.x + threadIdx.x;
    if (i < E) {
        row32[i] = (int)ei[i];
        col32[i] = (int)ei[(long)E + i];
    }
}

// deg[col] += w for each edge, + 1 for each self-loop
__global__ void k_degree(const int* __restrict__ col32, const float* __restrict__ ew,
                         float* __restrict__ deg, int E, int N) {
    long i = (long)blockIdx.x * blockDim.x + threadIdx.x;
    if (i < E) {
        atomicAdd(&deg[col32[i]], ew[i]);
    } else if (i < (long)E + N) {
        atomicAdd(&deg[i - E], 1.0f);
    }
}

__global__ void k_rsqrt_deg(float* __restrict__ deg, int N) {
    int i = blockIdx.x * blockDim.x + threadIdx.x;
    if (i < N) {
        float d = deg[i];
        deg[i] = (d > 0.0f) ? rsqrtf(d) : 0.0f;
    }
}

// One-time: norm[e] = dis[row]*w*dis[col]  (reused by both GCN layers)
__global__ void k_norm(const int* __restrict__ row32, const int* __restrict__ col32,
                       const float* __restrict__ ew, const float* __restrict__ dis,
                       float* __restrict__ norm, int E) {
    long i = (long)blockIdx.x * blockDim.x + threadIdx.x;
    if (i < E) norm[i] = dis[row32[i]] * ew[i] * dis[col32[i]];
}

// ---------------- fp32 WMMA GEMM: H[M,NC] = X[M,K] @ W[K,NC] ----------------
// Block = 256 threads = 8 waves; each wave owns a 16-row tile, full NC width.
// M must be a multiple of 128; K a multiple of 4; NC a multiple of 16.
template <int K, int NC>
__global__ void __launch_bounds__(256) k_gemm_wmma(const float* __restrict__ X,
                                                   const float* __restrict__ W,
                                                   float* __restrict__ H) {
    constexpr int NT = NC / 16;
    const int lane = threadIdx.x & 31;
    const int wave = threadIdx.x >> 5;
    const long row = (long)blockIdx.x * 128 + (long)wave * 16;
    const int lm = lane & 15;   // M (A) / N (B,C)
    const int lh = lane >> 4;   // K-half selector

    v8f acc[NT];
    v8f zinit = {};
#pragma unroll
    for (int t = 0; t < NT; ++t) acc[t] = zinit;

    const float* __restrict__ xrow = X + (row + lm) * K;

    for (int k = 0; k < K; k += 4) {
        // A fragment: lane holds row M=lm, cols k+2*lh, k+2*lh+1 (ISA 7.12.2)
        v2f a;
        a.x = xrow[k + 2 * lh];
        a.y = xrow[k + 2 * lh + 1];
        const float* __restrict__ w0 = W + (long)(k + 2 * lh) * NC + lm;
        const float* __restrict__ w1 = W + (long)(k + 2 * lh + 1) * NC + lm;
#pragma unroll
        for (int t = 0; t < NT; ++t) {
            v2f b;
            b.x = w0[t * 16];
            b.y = w1[t * 16];
            acc[t] = __builtin_amdgcn_wmma_f32_16x16x4_f32(
                false, a, false, b, (short)0, acc[t], false, false);
        }
    }

    // D layout: VGPR v -> M = v + 8*lh, N = lm
#pragma unroll
    for (int t = 0; t < NT; ++t) {
#pragma unroll
        for (int v = 0; v < 8; ++v) {
            H[(row + v + 8 * lh) * NC + t * 16 + lm] = acc[t][v];
        }
    }
}

// ---------------- edge scatter: A[col] += H[row] * norm[e] ----------------
// `chunks` consecutive lanes share one edge: metadata loads broadcast-collapse,
// float4 gather + 4 atomics are coalesced across the feature row.
__global__ void k_scatter(const int* __restrict__ row32, const int* __restrict__ col32,
                          const float* __restrict__ norm, const float* __restrict__ H,
                          float* __restrict__ A, int E, int NC) {
    const int chunks = NC >> 2;
    long idx = (long)blockIdx.x * blockDim.x + threadIdx.x;
    if (idx >= (long)E * chunks) return;
    int e = (int)(idx / chunks);
    int c = (int)(idx - (long)e * chunks) << 2;
    long r = (long)row32[e];
    long col = (long)col32[e];
    float nrm = norm[e];
    const float4 h = *(const float4*)(H + r * NC + c);
    float* o = A + col * NC + c;
    atomicAdd(o + 0, h.x * nrm);
    atomicAdd(o + 1, h.y * nrm);
    atomicAdd(o + 2, h.z * nrm);
    atomicAdd(o + 3, h.w * nrm);
}

// A = mish(A + H*dis^2 + b)   (self-loop + bias + activation, fused)
__global__ void k_self_bias_mish(const float* __restrict__ dis, const float* __restrict__ H,
                                 const float* __restrict__ b, float* __restrict__ A,
                                 long N, int NC) {
    long idx = (long)blockIdx.x * blockDim.x + threadIdx.x;
    if (idx >= N * NC) return;
    long i = idx / NC;
    int c = (int)(idx - i * NC);
    float d = dis[i];
    A[idx] = mishf(A[idx] + H[idx] * d * d + b[c]);
}

// feat = mish(A[N,64] @ Wro[64,8] + bro)
__global__ void k_readout(const float* __restrict__ A, const float* __restrict__ Wro,
                          const float* __restrict__ bro, float* __restrict__ feat, long N) {
    long idx = (long)blockIdx.x * blockDim.x + threadIdx.x;
    if (idx >= N * 8) return;
    long i = idx >> 3;
    int o = (int)(idx & 7);
    const float* a = A + i * 64;
    float s = bro[o];
#pragma unroll
    for (int k = 0; k < 64; ++k) s += a[k] * Wro[k * 8 + o];
    feat[idx] = mishf(s);
}

// z[B,200] = feat[B,1600] @ Wfc1[1600,200] + bfc1
__global__ void k_fc1(const float* __restrict__ feat, const float* __restrict__ W,
                      const float* __restrict__ b, float* __restrict__ z, int B) {
    int idx = blockIdx.x * blockDim.x + threadIdx.x;
    if (idx >= B * NUM_ROI) return;
    int r = idx / NUM_ROI;
    int c = idx - r * NUM_ROI;
    const float* f = feat + (long)r * (NUM_ROI * 8);
    float s = b[c];
    for (int k = 0; k < NUM_ROI * 8; ++k) s += f[k] * W[(long)k * NUM_ROI + c];
    z[idx] = s;
}

// BatchNorm (training mode, biased var) + mish. One block per feature column.
__global__ void __launch_bounds__(256) k_bn_mish(const float* __restrict__ z,
                                                 const float* __restrict__ gamma,
                                                 const float* __restrict__ beta,
                                                 float* __restrict__ mid, int B) {
    int c = blockIdx.x;    // 0..199
    int r = threadIdx.x;   // 0..255
    __shared__ float s1[256];
    __shared__ float s2[256];
    float v = (r < B) ? z[(long)r * NUM_ROI + c] : 0.0f;
    s1[r] = v;
    s2[r] = v * v;
    __syncthreads();
    for (int off = 128; off > 0; off >>= 1) {
        if (r < off) { s1[r] += s1[r + off]; s2[r] += s2[r + off]; }
        __syncthreads();
    }
    float mu = s1[0] / (float)B;
    float var = s2[0] / (float)B - mu * mu;
    if (r < B)
        mid[(long)r * NUM_ROI + c] =
            mishf((v - mu) * rsqrtf(var + 1e-5f) * gamma[c] + beta[c]);
}

// logits = mid @ Wfc2 + bfc2 ; cls = relu(mid @ Wd1 + bd1) @ Wd2 + bd2
// One block per batch row. Output: [B*2 logits][B*6 cls] flat.
__global__ void __launch_bounds__(256) k_heads(const float* __restrict__ mid,
        const float* __restrict__ Wfc2, const float* __restrict__ bfc2,
        const float* __restrict__ Wd1, const float* __restrict__ bd1,
        const float* __restrict__ Wd2, const float* __restrict__ bd2,
        float* __restrict__ out, int B) {
    int r = blockIdx.x;
    int t = threadIdx.x;
    __shared__ float m[NUM_ROI];
    __shared__ float tv[64];
    if (t < NUM_ROI) m[t] = mid[(long)r * NUM_ROI + t];
    __syncthreads();
    if (t < 2) {
        float s = bfc2[t];
        for (int k = 0; k < NUM_ROI; ++k) s += m[k] * Wfc2[k * 2 + t];
        out[r * 2 + t] = s;
    }
    if (t < 64) {
        float s = bd1[t];
        for (int k = 0; k < NUM_ROI; ++k) s += m[k] * Wd1[k * 64 + t];
        tv[t] = fmaxf(s, 0.0f);
    }
    __syncthreads();
    if (t < 6) {
        float s = bd2[t];
#pragma unroll
        for (int k = 0; k < 64; ++k) s += tv[k] * Wd2[k * 6 + t];
        out[(long)B * 2 + r * 6 + t] = s;
    }
}

extern "C" void kernel_launch(void* const* d_in, const int* in_sizes, int n_in,
                              void* d_out, int out_size, void* d_ws, size_t ws_size,
                              hipStream_t stream) {
    const float*     x    = (const float*)d_in[0];
    const long long* ei   = (const long long*)d_in[1];
    const float*     ea   = (const float*)d_in[2];
    const float* W1 = (const float*)d_in[3];  const float* b1  = (const float*)d_in[4];
    const float* W2 = (const float*)d_in[5];  const float* b2  = (const float*)d_in[6];
    const float* Wro = (const float*)d_in[7]; const float* bro = (const float*)d_in[8];
    const float* Wfc1 = (const float*)d_in[9]; const float* bfc1 = (const float*)d_in[10];
    const float* gamma = (const float*)d_in[11]; const float* beta = (const float*)d_in[12];
    const float* Wfc2 = (const float*)d_in[13]; const float* bfc2 = (const float*)d_in[14];
    const float* Wd1 = (const float*)d_in[15]; const float* bd1 = (const float*)d_in[16];
    const float* Wd2 = (const float*)d_in[17]; const float* bd2 = (const float*)d_in[18];
    float* out = (float*)d_out;

    const long N = (long)in_sizes[0] / NUM_ROI;  // 51200
    const int  E = in_sizes[2];                  // 1638400
    const int  B = (int)(N / NUM_ROI);           // 256

    // Workspace layout (floats / ints)
    float* ws    = (float*)d_ws;
    float* deg   = ws;                          // N      (becomes dis = rsqrt(deg))
    float* H     = deg + N;                     // N*128
    float* A     = H + N * 128;                 // N*128
    float* feat  = A + N * 128;                 // N*8
    float* z     = feat + N * 8;                // B*200
    float* mid   = z + (long)B * NUM_ROI;       // B*200
    float* norm  = mid + (long)B * NUM_ROI;     // E
    int*   row32 = (int*)(norm + E);            // E
    int*   col32 = row32 + E;                   // E

    const int TPB = 256;
    auto blocks = [](long n, int tpb) { return (int)((n + tpb - 1) / tpb); };

    // ---- one-time edge prep: int32 indices + fused edge norms ----
    k_prep_idx<<<blocks(E, TPB), TPB, 0, stream>>>(ei, row32, col32, E);
    k_zero<<<blocks(N, TPB), TPB, 0, stream>>>(deg, N);
    k_degree<<<blocks((long)E + N, TPB), TPB, 0, stream>>>(col32, ea, deg, E, (int)N);
    k_rsqrt_deg<<<blocks(N, TPB), TPB, 0, stream>>>(deg, (int)N);
    k_norm<<<blocks(E, TPB), TPB, 0, stream>>>(row32, col32, ea, deg, norm, E);

    // ---- layer 1: H = X @ W1 (WMMA fp32), aggregate, mish ----
    k_gemm_wmma<NUM_ROI, 128><<<(int)(N / 128), 256, 0, stream>>>(x, W1, H);
    k_zero<<<blocks(N * 128, TPB), TPB, 0, stream>>>(A, N * 128);
    k_scatter<<<blocks((long)E * 32, TPB), TPB, 0, stream>>>(row32, col32, norm, H, A, E, 128);
    k_self_bias_mish<<<blocks(N * 128, TPB), TPB, 0, stream>>>(deg, H, b1, A, N, 128);

    // ---- layer 2: H = A @ W2 (WMMA fp32), aggregate, mish ----
    k_gemm_wmma<128, 64><<<(int)(N / 128), 256, 0, stream>>>(A, W2, H);
    k_zero<<<blocks(N * 64, TPB), TPB, 0, stream>>>(A, N * 64);
    k_scatter<<<blocks((long)E * 16, TPB), TPB, 0, stream>>>(row32, col32, norm, H, A, E, 64);
    k_self_bias_mish<<<blocks(N * 64, TPB), TPB, 0, stream>>>(deg, H, b2, A, N, 64);

    // ---- readout + heads ----
    k_readout<<<blocks(N * 8, TPB), TPB, 0, stream>>>(A, Wro, bro, feat, N);
    k_fc1<<<blocks((long)B * NUM_ROI, TPB), TPB, 0, stream>>>(feat, Wfc1, bfc1, z, B);
    k_bn_mish<<<NUM_ROI, 256, 0, stream>>>(z, gamma, beta, mid, B);
    k_heads<<<B, 256, 0, stream>>>(mid, Wfc2, bfc2, Wd1, bd1, Wd2, bd2, out, B);
}